// BoundaryPredictor2_55551107006574
// MI455X (gfx1250) — compile-verified
//
#include <hip/hip_runtime.h>
#include <math.h>

// ---------------------------------------------------------------------------
// BoundaryPredictor2 forward for MI455X (gfx1250, wave32, WMMA + TDM)
// B=8, L=4096, D=1024.
// cos[b,p] = rn[p]*rn[p+1] * ( h[p]^T (Wq^T Wk) h[p+1] )   (bilinear trick)
// ---------------------------------------------------------------------------

#define BB 8
#define LL 4096
#define DD 1024
#define PP 4095                      // pairs per batch row (L-1)
#define OUT_POOL 0
#define OUT_LOSS 33554432            // B*L*D
#define OUT_NB   33554433
#define OUT_TP   33554434
#define OUT_VALID 33554435

typedef float v2f __attribute__((ext_vector_type(2)));
typedef float v8f __attribute__((ext_vector_type(8)));
typedef unsigned int v4u __attribute__((ext_vector_type(4)));
typedef unsigned int v8u __attribute__((ext_vector_type(8)));

// ---------------------------------------------------------------------------
// Kernel 1: M = Wq^T @ Wk (1024x1024, fp32 WMMA 16x16x4), stored K-pair
// interleaved:  Mp[(i>>1)*2048 + j*2 + (i&1)]  so k3's B fragment is one b64.
// grid 512 x 256 threads (8 waves), one 16x16 tile per wave.
// ---------------------------------------------------------------------------
__global__ void bp2_k1_wqwk(const float* __restrict__ Wq,
                            const float* __restrict__ Wk,
                            float* __restrict__ Mp) {
  const int tid  = threadIdx.x;
  const int lane = tid & 31;
  const int wave = tid >> 5;
  const int half = lane >> 4;
  const int lr   = lane & 15;
  const int tile = blockIdx.x * 8 + wave;   // 0..4095
  const int i0 = (tile >> 6) * 16;
  const int j0 = (tile & 63) * 16;

  v8f c = {0.f,0.f,0.f,0.f,0.f,0.f,0.f,0.f};
  for (int k = 0; k < DD; k += 4) {
    const int ka = k + 2 * half;
    v2f a, b;
    // A[M][K] = Wq^T[i0+M][K] = Wq[K][i0+M]
    a.x = Wq[(size_t)ka * DD + i0 + lr];
    a.y = Wq[(size_t)(ka + 1) * DD + i0 + lr];
    // B[K][N] = Wk[K][j0+N]
    b.x = Wk[(size_t)ka * DD + j0 + lr];
    b.y = Wk[(size_t)(ka + 1) * DD + j0 + lr];
    c = __builtin_amdgcn_wmma_f32_16x16x4_f32(false, a, false, b,
                                              (short)0, c, false, false);
  }
#pragma unroll
  for (int r = 0; r < 8; ++r) {
    const int i = i0 + r + 8 * half;
    Mp[(size_t)(i >> 1) * 2048 + (size_t)(j0 + lr) * 2 + (i & 1)] = c[r];
  }
}

// ---------------------------------------------------------------------------
// Kernel 2: reciprocal L2 norms of every hidden row: rn = 1/max(||h||,1e-12)
// ---------------------------------------------------------------------------
__global__ void bp2_k2_rnorm(const float* __restrict__ hidden,
                             float* __restrict__ rn) {
  __shared__ float red[256];
  const int t = threadIdx.x;
  const float4* hp = (const float4*)(hidden + (size_t)blockIdx.x * DD);
  float4 v = hp[t];
  red[t] = v.x * v.x + v.y * v.y + v.z * v.z + v.w * v.w;
  __syncthreads();
  for (int off = 128; off > 0; off >>= 1) {
    if (t < off) red[t] += red[t + off];
    __syncthreads();
  }
  if (t == 0) rn[blockIdx.x] = 1.0f / fmaxf(sqrtf(red[0]), 1e-12f);
}

// ---------------------------------------------------------------------------
// Kernel 3: cos via WMMA, raw 16x512 hidden tile staged to LDS by the Tensor
// Data Mover (2D D#, pad 1 DWORD / 128 DWORDs -> LDS row stride 516 floats,
// bank-conflict-free A-fragment reads).  One workgroup per 16-pair tile,
// 8 waves split 64 j-tiles; deterministic shfl_xor + LDS reduction.
// ---------------------------------------------------------------------------
__global__ void bp2_k3_cos(const float* __restrict__ hidden,
                           const float* __restrict__ rn,
                           const float* __restrict__ Mp,
                           float* __restrict__ cosws) {
  const int b    = blockIdx.x >> 8;      // 256 tiles per batch
  const int tile = blockIdx.x & 255;
  const int pp   = tile * 16;
  const int tid  = threadIdx.x;
  const int lane = tid & 31;
  const int wave = tid >> 5;
  const int half = lane >> 4;
  const int lr   = lane & 15;

  __shared__ float n1t[16 * 516];        // [m][k + (k>>7)]  (TDM pad layout)
  __shared__ float partial[16 * 8];      // [m][wave]

  float accM[8];
#pragma unroll
  for (int r = 0; r < 8; ++r) accM[r] = 0.f;

  for (int kc = 0; kc < DD; kc += 512) {
    // --- stage raw hidden tile (16 rows x 512 cols) via TDM ---------------
    if (wave == 0) {
      const unsigned long long ga =
          (unsigned long long)(const void*)(hidden + (size_t)(b * LL + pp) * DD + kc);
      const unsigned ldsb = (unsigned)(unsigned long long)(void*)&n1t[0];
      v4u g0;
      g0[0] = 1u;                                   // count=1, user mode
      g0[1] = ldsb;                                 // lds_addr
      g0[2] = (unsigned)ga;                         // global_addr[31:0]
      g0[3] = (unsigned)((ga >> 32) & 0x01FFFFFFu)  // global_addr[56:32]
              | (2u << 30);                         // type=2 (image)
      v8u g1;
      g1[0] = (2u << 16)      // data_size = 4B
            | (1u << 20)      // pad_enable
            | (6u << 22)      // pad_interval: 128 DWORDs
            | (0u << 25);     // pad_amount: 1 DWORD
      g1[1] = (512u << 16);   // tensor_dim0 = 512
      g1[2] = (16u << 16);    // tensor_dim1 = 16
      g1[3] = (512u << 16);   // tile_dim0 = 512
      g1[4] = 16u;            // tile_dim1 = 16 (tile_dim2 = 0)
      g1[5] = 1024u;          // tensor_dim0_stride = D elements
      g1[6] = 0u;
      g1[7] = 0u;
      asm volatile("tensor_load_to_lds %0, %1" :: "s"(g0), "s"(g1) : "memory");
      __builtin_amdgcn_s_wait_tensorcnt(0);
    }
    __syncthreads();

    for (int jt = wave; jt < 64; jt += 8) {
      const int j0 = jt * 16;
      v8f c = {0.f,0.f,0.f,0.f,0.f,0.f,0.f,0.f};
      for (int kk = 0; kk < 512; kk += 4) {
        const int ka = kk + 2 * half;                    // even
        const int idxA = lr * 516 + ka + (ka >> 7);      // TDM pad offset
        v2f a;
        a.x = n1t[idxA];
        a.y = n1t[idxA + 1];
        // B fragment: rows (kc+ka, kc+ka+1), col j0+lr -> one contiguous b64
        const v2f bf = *(const v2f*)(Mp + (size_t)((kc + ka) >> 1) * 2048 +
                                     (size_t)(j0 + lr) * 2);
        c = __builtin_amdgcn_wmma_f32_16x16x4_f32(false, a, false, bf,
                                                  (short)0, c, false, false);
      }
      // fold C against raw h2 (row-wise dot): element (M=r+8*half, N=lr)
#pragma unroll
      for (int r = 0; r < 8; ++r) {
        const int m = r + 8 * half;
        const int p = pp + m;
        if (p < PP) {
          const float h2 = hidden[(size_t)(b * LL + p + 1) * DD + j0 + lr];
          accM[r] += c[r] * h2;
        }
      }
    }
    __syncthreads();
  }

  // reduce 16 lanes of each half, then 8 waves (deterministic order)
#pragma unroll
  for (int r = 0; r < 8; ++r) {
    float v = accM[r];
    v += __shfl_xor(v, 1, 32);
    v += __shfl_xor(v, 2, 32);
    v += __shfl_xor(v, 4, 32);
    v += __shfl_xor(v, 8, 32);
    if (lr == 0) partial[(r + 8 * half) * 8 + wave] = v;
  }
  __syncthreads();
  if (tid < 16) {
    float s = 0.f;
#pragma unroll
    for (int w = 0; w < 8; ++w) s += partial[tid * 8 + w];
    const int p = pp + tid;
    if (p < PP)
      cosws[b * LL + p] = s * rn[b * LL + p] * rn[b * LL + p + 1];
  }
}

// ---------------------------------------------------------------------------
// Kernel 4: hard bits + per-row scan -> segstart scatter, counts, mask sum.
// ---------------------------------------------------------------------------
__global__ void bp2_k4_scan(const float* __restrict__ mask,
                            const float* __restrict__ noise,
                            const float* __restrict__ cosws,
                            int* __restrict__ segstart,
                            int* __restrict__ counts,
                            float* __restrict__ masksum) {
  const int b = blockIdx.x;
  const int t = threadIdx.x;
  __shared__ int firstpad_s;
  __shared__ int sums[256];
  __shared__ float msum_s[256];
  if (t == 0) firstpad_s = LL;
  __syncthreads();

  const int l0 = t * 16;
  int localfp = LL;
  float mloc = 0.f;
  for (int i = 0; i < 16; ++i) {
    const float m = mask[b * LL + l0 + i];
    if (m == 0.f && (l0 + i) < localfp) localfp = l0 + i;
    mloc += m;
  }
  if (localfp < LL) atomicMin(&firstpad_s, localfp);
  msum_s[t] = mloc;
  __syncthreads();
  const int fp = firstpad_s;

  unsigned bits = 0;
  int lsum = 0;
  for (int i = 0; i < 16; ++i) {
    const int l = l0 + i;
    const float m = mask[b * LL + l];
    int bit;
    if (l == 0) {
      bit = 1;  // probs[:,0]=1 -> logit(1-1e-7)+logistic > 0 always
    } else {
      float p = 0.5f * (1.0f - cosws[b * LL + (l - 1)]);
      p = fminf(fmaxf(p, 0.f), 1.f);
      p = fminf(fmaxf(p, 1e-7f), 1.f - 1e-7f);
      const float logit = logf(p) - log1pf(-p);
      float u = noise[b * LL + l];
      u = fminf(fmaxf(u, 1e-6f), 1.f - 1e-6f);
      const float lg = logf(u) - log1pf(-u);
      bit = ((logit + lg) > 0.f) ? 1 : 0;
    }
    if (m == 0.f) bit = 0;
    if (fp < LL && l == fp - 1) bit = 1;  // force boundary at last real token
    bits |= ((unsigned)bit) << i;
    lsum += bit;
  }
  sums[t] = lsum;
  __syncthreads();
  for (int off = 1; off < 256; off <<= 1) {
    const int mine = sums[t];
    const int add = (t >= off) ? sums[t - off] : 0;
    __syncthreads();
    sums[t] = mine + add;
    __syncthreads();
  }
  int run = sums[t] - lsum;  // exclusive prefix
  for (int i = 0; i < 16; ++i) {
    if ((bits >> i) & 1u) {
      run += 1;
      segstart[b * LL + (run - 1)] = l0 + i;
    }
  }
  if (t == 255) counts[b] = sums[255];
  __syncthreads();
  for (int off = 128; off > 0; off >>= 1) {
    if (t < off) msum_s[t] += msum_s[t + off];
    __syncthreads();
  }
  if (t == 0) masksum[b] = msum_s[0];
}

// ---------------------------------------------------------------------------
// Kernel 5: mean-pool contiguous segments -> pooled (zeros beyond counts).
// ---------------------------------------------------------------------------
__global__ void bp2_k5_pool(const float* __restrict__ hidden,
                            const int* __restrict__ segstart,
                            const int* __restrict__ counts,
                            float* __restrict__ out) {
  const int b = blockIdx.x >> 12;
  const int s = blockIdx.x & 4095;
  const int t = threadIdx.x;
  float4* dst = (float4*)out + ((size_t)(b * LL + s) * 256) + t;
  const int cnt = counts[b];
  if (s >= cnt) {
    float4 z; z.x = 0.f; z.y = 0.f; z.z = 0.f; z.w = 0.f;
    *dst = z;
    return;
  }
  const int start = segstart[b * LL + s];
  const int end = (s + 1 < cnt) ? segstart[b * LL + s + 1] : LL;
  float4 acc; acc.x = 0.f; acc.y = 0.f; acc.z = 0.f; acc.w = 0.f;
  for (int l = start; l < end; ++l) {
    const float4 v = *((const float4*)hidden + ((size_t)(b * LL + l) * 256) + t);
    acc.x += v.x; acc.y += v.y; acc.z += v.z; acc.w += v.w;
  }
  const float inv = 1.0f / (float)(end - start);
  acc.x *= inv; acc.y *= inv; acc.z *= inv; acc.w *= inv;
  *dst = acc;
}

// ---------------------------------------------------------------------------
// Kernel 6: scalars (binomial NLL loss, num_boundaries, total_positions)
// ---------------------------------------------------------------------------
__global__ void bp2_k6_final(const int* __restrict__ counts,
                             const float* __restrict__ masksum,
                             float* __restrict__ out) {
  if (threadIdx.x == 0 && blockIdx.x == 0) {
    float nb = 0.f, tp = 0.f;
    for (int b = 0; b < BB; ++b) { nb += (float)counts[b]; tp += masksum[b]; }
    const float n = tp, k = nb;
    const float lp = lgammaf(n + 1.f) - lgammaf(k + 1.f) - lgammaf(n - k + 1.f)
                   + k * logf(0.2f) + (n - k) * log1pf(-0.2f);
    out[OUT_LOSS] = -lp / 4096.0f;
    out[OUT_NB] = nb;
    out[OUT_TP] = tp;
  }
}

// ---------------------------------------------------------------------------
// Kernel 7: valid mask output
// ---------------------------------------------------------------------------
__global__ void bp2_k7_valid(const int* __restrict__ counts,
                             float* __restrict__ out) {
  const int i = blockIdx.x * 256 + threadIdx.x;  // 0..32767
  const int b = i >> 12;
  const int j = i & 4095;
  out[OUT_VALID + i] = (j < counts[b]) ? 1.0f : 0.0f;
}

// ---------------------------------------------------------------------------
extern "C" void kernel_launch(void* const* d_in, const int* in_sizes, int n_in,
                              void* d_out, int out_size, void* d_ws, size_t ws_size,
                              hipStream_t stream) {
  const float* hidden = (const float*)d_in[0];
  const float* mask   = (const float*)d_in[1];
  const float* noise  = (const float*)d_in[2];
  const float* Wq     = (const float*)d_in[3];
  const float* Wk     = (const float*)d_in[4];
  float* out = (float*)d_out;
  float* ws  = (float*)d_ws;

  // workspace layout (float units): ~4.6 MB total
  float* Mp       = ws;                       // 1048576 (K-pair interleaved M)
  float* rn       = ws + 1048576;             // 32768
  float* cosws    = ws + 1081344;             // 32768
  int*   segstart = (int*)(ws + 1114112);     // 32768
  int*   counts   = (int*)(ws + 1146880);     // 8
  float* msum     = ws + 1146888;             // 8

  bp2_k1_wqwk <<<512,   256, 0, stream>>>(Wq, Wk, Mp);
  bp2_k2_rnorm<<<32768, 256, 0, stream>>>(hidden, rn);
  bp2_k3_cos  <<<2048,  256, 0, stream>>>(hidden, rn, Mp, cosws);
  bp2_k4_scan <<<8,     256, 0, stream>>>(mask, noise, cosws, segstart, counts, msum);
  bp2_k5_pool <<<32768, 256, 0, stream>>>(hidden, segstart, counts, out);
  bp2_k6_final<<<1,     32,  0, stream>>>(counts, msum, out);
  bp2_k7_valid<<<128,   256, 0, stream>>>(counts, out);
}